// AttnAggLayer_84052509983411
// MI455X (gfx1250) — compile-verified
//
#include <hip/hip_runtime.h>

#define N_NODES 100000
#define N_EDGES 1600000
#define N_TEAMS 1024
// IN_DIM = EMB_DIM = 64, NTF = 32, concat = 96

typedef __attribute__((ext_vector_type(2))) float v2f;
typedef __attribute__((ext_vector_type(8))) float v8f;

// ---- monotone float <-> uint mapping so atomicMax(u32) == float max ----
__device__ __forceinline__ unsigned f2mono(float f) {
    unsigned u = __float_as_uint(f);
    return (u & 0x80000000u) ? ~u : (u | 0x80000000u);
}
__device__ __forceinline__ float mono2f(unsigned u) {
    return (u & 0x80000000u) ? __uint_as_float(u & 0x7fffffffu)
                             : __uint_as_float(~u);
}

// ---- zero-init workspace region (z, m, denom, team_sum, cnt) ----
__global__ void init_zero_kernel(unsigned* __restrict__ p, size_t n) {
    size_t i = (size_t)blockIdx.x * blockDim.x + threadIdx.x;
    if (i < n) p[i] = 0u;
}

// ---- per-node: a_src = f . attn_w[:64], a_dst = f . attn_w[64:], team count ----
__global__ void node_kernel(const float* __restrict__ f,
                            const float* __restrict__ attn_w,
                            const int* __restrict__ node_team,
                            float* __restrict__ a_src,
                            float* __restrict__ a_dst,
                            unsigned* __restrict__ cnt) {
    int i = blockIdx.x * blockDim.x + threadIdx.x;
    if (i >= N_NODES) return;
    const float4* fr  = (const float4*)(f + (size_t)i * 64);
    const float4* ws  = (const float4*)(attn_w);       // [0..63]
    const float4* wd  = (const float4*)(attn_w + 64);  // [64..127]
    float s = 0.f, d = 0.f;
#pragma unroll
    for (int k = 0; k < 16; ++k) {
        float4 v = fr[k], a = ws[k], b = wd[k];
        s += v.x * a.x + v.y * a.y + v.z * a.z + v.w * a.w;
        d += v.x * b.x + v.y * b.y + v.z * b.z + v.w * b.w;
    }
    a_src[i] = s;
    a_dst[i] = d;
    atomicAdd(&cnt[node_team[i]], 1u);
}

__device__ __forceinline__ float edge_logit(const float* a_src, const float* a_dst,
                                            int s, int d) {
    float e = a_src[s] + a_dst[d];
    return e > 0.f ? e : 0.01f * e;   // leaky_relu(.., 0.01)
}

// ---- edge pass 1: segment max of logits by src ----
__global__ void edge_max_kernel(const int* __restrict__ edge_src,
                                const int* __restrict__ edge_dst,
                                const float* __restrict__ a_src,
                                const float* __restrict__ a_dst,
                                unsigned* __restrict__ m) {
    int i = blockIdx.x * blockDim.x + threadIdx.x;
    if (i >= N_EDGES) return;
    int s = edge_src[i], d = edge_dst[i];
    atomicMax(&m[s], f2mono(edge_logit(a_src, a_dst, s, d)));
}

// ---- edge pass 2: denom = segment_sum(exp(e - m)) ----
__global__ void edge_expsum_kernel(const int* __restrict__ edge_src,
                                   const int* __restrict__ edge_dst,
                                   const float* __restrict__ a_src,
                                   const float* __restrict__ a_dst,
                                   const unsigned* __restrict__ m,
                                   float* __restrict__ denom) {
    int i = blockIdx.x * blockDim.x + threadIdx.x;
    if (i >= N_EDGES) return;
    int s = edge_src[i], d = edge_dst[i];
    float e = edge_logit(a_src, a_dst, s, d);
    atomicAdd(&denom[s], __expf(e - mono2f(m[s])));
}

// ---- edge pass 3: z[s] += alpha * f[d] (one wave32 per edge, 2 dims/lane) ----
__global__ void edge_agg_kernel(const int* __restrict__ edge_src,
                                const int* __restrict__ edge_dst,
                                const float* __restrict__ a_src,
                                const float* __restrict__ a_dst,
                                const unsigned* __restrict__ m,
                                const float* __restrict__ denom,
                                const float* __restrict__ f,
                                float* __restrict__ z) {
    int gid  = blockIdx.x * blockDim.x + threadIdx.x;
    int edge = gid >> 5;          // wave32 per edge
    int lane = gid & 31;
    int s = edge_src[edge], d = edge_dst[edge];
    float e = edge_logit(a_src, a_dst, s, d);
    float alpha = __expf(e - mono2f(m[s])) / denom[s];
    const float2* frow = (const float2*)(f + (size_t)d * 64);
    float2 v = frow[lane];
    float* zr = z + (size_t)s * 64 + 2 * lane;
    atomicAdd(zr + 0, alpha * v.x);
    atomicAdd(zr + 1, alpha * v.y);
}

// ---- z' = elu(z @ fc_w^T) via V_WMMA_F32_16X16X4_F32, then team segment-sum ----
// Block = 64 threads (2 waves); each wave owns 16 node-rows x 64 emb cols.
__global__ __launch_bounds__(64) void fc_elu_team_kernel(
        const float* __restrict__ z,
        const float* __restrict__ fc_w,
        const int* __restrict__ node_team,
        float* __restrict__ team_sum) {
    __shared__ float ldsT[64 * 64];   // ldsT[k*64 + e] = fc_w[e*64 + k] = B[k][e]
    for (int i = threadIdx.x; i < 64 * 64; i += 64) {
        int e = i & 63, k = i >> 6;
        ldsT[i] = fc_w[e * 64 + k];
    }
    __syncthreads();

    int wave = blockIdx.x * 2 + (threadIdx.x >> 5);
    int lane = threadIdx.x & 31;
    int half = lane >> 4;             // 0: lanes 0-15, 1: lanes 16-31
    int l16  = lane & 15;
    int rowbase = wave * 16;          // 6250 waves * 16 rows == 100000 exactly

    // A-matrix 16x4 f32 layout: lanes 0-15 hold K={0,1}, lanes 16-31 hold K={2,3}
    const float* zrow = z + (size_t)(rowbase + l16) * 64 + 2 * half;

    v8f acc0 = {}, acc1 = {}, acc2 = {}, acc3 = {};
#pragma unroll
    for (int kb = 0; kb < 16; ++kb) {
        v2f a = *(const v2f*)(zrow + kb * 4);
        int krow = kb * 4 + 2 * half;
        const float* bp = &ldsT[krow * 64 + l16];   // B VGPR0: K=krow, VGPR1: K=krow+1
        v2f b0, b1, b2, b3;
        b0.x = bp[0];   b0.y = bp[64];      // cols  0..15
        b1.x = bp[16];  b1.y = bp[80];      // cols 16..31
        b2.x = bp[32];  b2.y = bp[96];      // cols 32..47
        b3.x = bp[48];  b3.y = bp[112];     // cols 48..63
        acc0 = __builtin_amdgcn_wmma_f32_16x16x4_f32(false, a, false, b0, (short)0, acc0, false, false);
        acc1 = __builtin_amdgcn_wmma_f32_16x16x4_f32(false, a, false, b1, (short)0, acc1, false, false);
        acc2 = __builtin_amdgcn_wmma_f32_16x16x4_f32(false, a, false, b2, (short)0, acc2, false, false);
        acc3 = __builtin_amdgcn_wmma_f32_16x16x4_f32(false, a, false, b3, (short)0, acc3, false, false);
    }

    // D layout: VGPR r -> row (r + 8*half), col = tile*16 + l16. ELU, then team scatter-add.
#pragma unroll
    for (int r = 0; r < 8; ++r) {
        int node = rowbase + r + 8 * half;
        int team = node_team[node];
        float* trow = team_sum + (size_t)team * 64 + l16;
        float v0 = acc0[r]; v0 = v0 > 0.f ? v0 : __expf(v0) - 1.0f;
        float v1 = acc1[r]; v1 = v1 > 0.f ? v1 : __expf(v1) - 1.0f;
        float v2 = acc2[r]; v2 = v2 > 0.f ? v2 : __expf(v2) - 1.0f;
        float v3 = acc3[r]; v3 = v3 > 0.f ? v3 : __expf(v3) - 1.0f;
        atomicAdd(trow + 0,  v0);
        atomicAdd(trow + 16, v1);
        atomicAdd(trow + 32, v2);
        atomicAdd(trow + 48, v3);
    }
}

// ---- final head: one wave32 per team; 96-dim dot + bias; also copy labels ----
__global__ void head_kernel(const float* __restrict__ team_sum,
                            const unsigned* __restrict__ cnt,
                            const float* __restrict__ team_features,
                            const float* __restrict__ out_w,
                            const float* __restrict__ out_b,
                            const float* __restrict__ team_labels,
                            float* __restrict__ out) {
    int gid  = blockIdx.x * blockDim.x + threadIdx.x;
    int team = gid >> 5;
    int lane = gid & 31;
    float c = fmaxf((float)cnt[team], 1.0f);
    float partial = 0.f;
#pragma unroll
    for (int j = 0; j < 3; ++j) {
        int col = lane + 32 * j;   // 0..95
        float x = (col < 64) ? team_sum[(size_t)team * 64 + col] / c
                             : team_features[(size_t)team * 32 + (col - 64)];
        partial += x * out_w[col];
    }
#pragma unroll
    for (int off = 16; off > 0; off >>= 1)
        partial += __shfl_xor(partial, off, 32);
    if (lane == 0) out[team] = partial + out_b[0];
    if (lane == 1) out[N_TEAMS + team] = team_labels[team];
}

extern "C" void kernel_launch(void* const* d_in, const int* in_sizes, int n_in,
                              void* d_out, int out_size, void* d_ws, size_t ws_size,
                              hipStream_t stream) {
    const float* f             = (const float*)d_in[0];
    const int*   edge_src      = (const int*)d_in[1];
    const int*   edge_dst      = (const int*)d_in[2];
    const int*   node_team     = (const int*)d_in[3];
    const float* team_features = (const float*)d_in[4];
    const float* team_labels   = (const float*)d_in[5];
    const float* attn_w        = (const float*)d_in[6];
    const float* fc_w          = (const float*)d_in[7];
    const float* out_w         = (const float*)d_in[8];
    const float* out_b         = (const float*)d_in[9];
    float* out = (float*)d_out;

    // workspace layout (~27.5 MB total)
    float*    z        = (float*)d_ws;                               // N*64
    unsigned* m        = (unsigned*)(z + (size_t)N_NODES * 64);      // N
    float*    denom    = (float*)(m + N_NODES);                      // N
    float*    team_sum = denom + N_NODES;                            // T*64
    unsigned* cnt      = (unsigned*)(team_sum + (size_t)N_TEAMS*64); // T
    float*    a_src    = (float*)(cnt + N_TEAMS);                    // N
    float*    a_dst    = a_src + N_NODES;                            // N

    size_t zero_elems = (size_t)N_NODES * 64 + 2 * (size_t)N_NODES
                      + (size_t)N_TEAMS * 64 + N_TEAMS;
    init_zero_kernel<<<(unsigned)((zero_elems + 255) / 256), 256, 0, stream>>>(
        (unsigned*)d_ws, zero_elems);

    node_kernel<<<(N_NODES + 255) / 256, 256, 0, stream>>>(
        f, attn_w, node_team, a_src, a_dst, cnt);

    edge_max_kernel<<<N_EDGES / 256, 256, 0, stream>>>(
        edge_src, edge_dst, a_src, a_dst, m);

    edge_expsum_kernel<<<N_EDGES / 256, 256, 0, stream>>>(
        edge_src, edge_dst, a_src, a_dst, m, denom);

    // one wave32 per edge: E*32 threads / 256 = E/8 blocks
    edge_agg_kernel<<<N_EDGES / 8, 256, 0, stream>>>(
        edge_src, edge_dst, a_src, a_dst, m, denom, f, z);

    // 6250 waves of 16 rows, 2 waves/block -> 3125 blocks (exact; EXEC all-ones)
    fc_elu_team_kernel<<<N_NODES / 32, 64, 0, stream>>>(
        z, fc_w, node_team, team_sum);

    head_kernel<<<(N_TEAMS * 32) / 256, 256, 0, stream>>>(
        team_sum, cnt, team_features, out_w, out_b, team_labels, out);
}